// TopKMoEAttention_15204184227982
// MI455X (gfx1250) — compile-verified
//
#include <hip/hip_runtime.h>
#include <math.h>

#define N_HEADS   16
#define N_EXPERTS 4
#define D_MODEL   1024
#define GATE_HID  128
#define B_        2
#define T_        1024
#define HD        64

typedef __attribute__((ext_vector_type(2))) float v2f;
typedef __attribute__((ext_vector_type(8))) float v8f;
typedef unsigned int u32x4 __attribute__((ext_vector_type(4)));
typedef int i32x4 __attribute__((ext_vector_type(4)));
typedef int i32x8 __attribute__((ext_vector_type(8)));

__device__ __forceinline__ v8f wmma_f32(v2f a, v2f b, v8f c) {
    // V_WMMA_F32_16X16X4_F32 : D = A(16x4) x B(4x16) + C(16x16), full fp32
    return __builtin_amdgcn_wmma_f32_16x16x4_f32(false, a, false, b, (short)0, c, false, false);
}

// ---------------------------------------------------------------------------
// Tensor Data Mover: async 2-D tile copy global -> LDS (CDNA5 ISA).
// d0_flags carries data_size + LDS pad config (group1 dword0).
// Tile exactly fits "tensor" (tensor_dim = tile_dim) so no OOB path triggers.
// ---------------------------------------------------------------------------
#define TDM_F32           (2u << 16)                       // data_size = 4B
#define TDM_F32_PAD16_4   ((2u << 16) | (1u << 20) | (3u << 22) | (3u << 25))
                          // +4 DWORD pad per 16 DWORDs -> LDS row stride 20/40
#define TDM_F32_PAD64_4   ((2u << 16) | (1u << 20) | (5u << 22) | (3u << 25))
                          // +4 DWORD pad per 64 DWORDs -> LDS row stride 68

__device__ __forceinline__ void tdm_load_2d(unsigned lds_addr, const void* gptr,
                                            unsigned tile_d0, unsigned tile_d1,
                                            unsigned stride_elems, unsigned d0_flags) {
    unsigned long long ga = (unsigned long long)(uintptr_t)gptr;
    u32x4 g0;
    g0[0] = 1u;                                            // count=1, user mode
    g0[1] = lds_addr;                                      // LDS byte offset
    g0[2] = (unsigned)(ga & 0xFFFFFFFFu);                  // global_addr[31:0]
    g0[3] = (unsigned)((ga >> 32) & 0x01FFFFFFu) | (2u << 30);  // addr[56:32] | type=2
    i32x8 g1;
    g1[0] = (int)d0_flags;                                 // wg_mask=0 | data_size | pad cfg
    g1[1] = (int)(tile_d0 << 16);                          // tensor_dim0[15:0] @ bits63:48
    g1[2] = (int)(tile_d1 << 16);                          // tensor_dim1[15:0] @ bits95:80
    g1[3] = (int)(tile_d0 << 16);                          // tile_dim0 @ bits127:112
    g1[4] = (int)(tile_d1 & 0xFFFF);                       // tile_dim1; tile_dim2=0 (2-D)
    g1[5] = (int)stride_elems;                             // tensor_dim0_stride[31:0]
    g1[6] = 0;                                             // stride hi / dim1_stride lo
    g1[7] = 0;
    i32x4 z4 = {0, 0, 0, 0};
#if __clang_major__ >= 23
    i32x8 z8 = {0, 0, 0, 0, 0, 0, 0, 0};
    __builtin_amdgcn_tensor_load_to_lds(g0, g1, z4, z4, z8, 0);
#else
    __builtin_amdgcn_tensor_load_to_lds(g0, g1, z4, z4, 0);
#endif
}

// ---------------------------------------------------------------------------
// Kernel 1: gating — pooled mean, 2-layer MLP, top-2 softmax -> weights[B][E]
// ---------------------------------------------------------------------------
__global__ __launch_bounds__(256)
void gate_kernel(const float* __restrict__ hs,
                 const float* __restrict__ Wg1, const float* __restrict__ bg1,
                 const float* __restrict__ Wg2, const float* __restrict__ bg2,
                 float* __restrict__ w_out) {
    __shared__ float pooled[B_ * D_MODEL];
    __shared__ float hid[B_ * GATE_HID];
    __shared__ float logits[B_ * N_EXPERTS];
    int t = threadIdx.x;
    for (int b = 0; b < B_; ++b)
        for (int c = t; c < D_MODEL; c += 256) {
            float s = 0.f;
            const float* p = hs + (size_t)b * T_ * D_MODEL + c;
            for (int si = 0; si < T_; ++si) s += p[(size_t)si * D_MODEL];
            pooled[b * D_MODEL + c] = s * (1.0f / (float)T_);
        }
    __syncthreads();
    if (t < GATE_HID) {
        for (int b = 0; b < B_; ++b) {
            float acc = bg1[t];
            const float* pb = pooled + b * D_MODEL;
            for (int c = 0; c < D_MODEL; ++c) acc += pb[c] * Wg1[c * GATE_HID + t];
            hid[b * GATE_HID + t] = fmaxf(acc, 0.f);
        }
    }
    __syncthreads();
    if (t < B_ * N_EXPERTS) {
        int b = t / N_EXPERTS, e = t % N_EXPERTS;
        float acc = bg2[e];
        for (int j = 0; j < GATE_HID; ++j) acc += hid[b * GATE_HID + j] * Wg2[j * N_EXPERTS + e];
        logits[t] = acc;
    }
    __syncthreads();
    if (t < B_) {
        int b = t;
        int i1 = 0; float v1 = logits[b * N_EXPERTS];
        for (int e = 1; e < N_EXPERTS; ++e) {
            float v = logits[b * N_EXPERTS + e];
            if (v > v1) { v1 = v; i1 = e; }
        }
        int i2 = -1; float v2 = -INFINITY;
        for (int e = 0; e < N_EXPERTS; ++e) {
            if (e == i1) continue;
            float v = logits[b * N_EXPERTS + e];
            if (v > v2) { v2 = v; i2 = e; }
        }
        float e2 = __expf(v2 - v1);
        float den = 1.f + e2;
        float w1 = 1.f / den, w2 = e2 / den;
        for (int e = 0; e < N_EXPERTS; ++e)
            w_out[b * N_EXPERTS + e] = (e == i1) ? w1 : (e == i2) ? w2 : 0.f;
    }
}

// ---------------------------------------------------------------------------
// Kernel 2: collapse experts  Wc[p][b] = sum_e w[b,e] * W_p[e]   (and biases)
// ---------------------------------------------------------------------------
__global__ void combine_w_kernel(const float* __restrict__ Wq, const float* __restrict__ Wk,
                                 const float* __restrict__ Wv, const float* __restrict__ wg,
                                 float* __restrict__ Wc) {
    const size_t CC = (size_t)D_MODEL * D_MODEL;
    const size_t total = 3ull * B_ * CC;
    for (size_t i = (size_t)blockIdx.x * blockDim.x + threadIdx.x; i < total;
         i += (size_t)gridDim.x * blockDim.x) {
        size_t z = i / CC, o = i % CC;
        int p = (int)(z / B_), b = (int)(z % B_);
        const float* W = (p == 0) ? Wq : (p == 1) ? Wk : Wv;
        float acc = 0.f;
#pragma unroll
        for (int e = 0; e < N_EXPERTS; ++e) acc += wg[b * N_EXPERTS + e] * W[e * CC + o];
        Wc[i] = acc;
    }
}

__global__ void combine_b_kernel(const float* __restrict__ bq, const float* __restrict__ bk,
                                 const float* __restrict__ bv, const float* __restrict__ wg,
                                 float* __restrict__ bc) {
    int total = 3 * B_ * D_MODEL;
    for (int i = blockIdx.x * blockDim.x + threadIdx.x; i < total;
         i += gridDim.x * blockDim.x) {
        int z = i / D_MODEL, c = i % D_MODEL;
        int p = z / B_, b = z % B_;
        const float* bs = (p == 0) ? bq : (p == 1) ? bk : bv;
        float acc = 0.f;
#pragma unroll
        for (int e = 0; e < N_EXPERTS; ++e) acc += wg[b * N_EXPERTS + e] * bs[e * D_MODEL + c];
        bc[i] = acc;
    }
}

// ---------------------------------------------------------------------------
// Kernel 3: tiled WMMA-f32 GEMM  Y[z] = X[z] @ W[z] + bias[z]
// 8 wave32, tile 64x64, BK=32. TDM double-buffered staging: wave0 DMAs the X
// tile (D# padding -> LDS row 40), wave1 DMAs the W tile; TENSORcnt + block
// barrier publish; next stage's DMA overlaps current stage's 16 WMMAs/wave.
// ---------------------------------------------------------------------------
__global__ __launch_bounds__(256)
void gemm_moe_kernel(const float* __restrict__ X, size_t x_zstride, int x_mod,
                     const float* __restrict__ W, size_t w_zstride,
                     const float* __restrict__ bias, size_t b_zstride,
                     float* __restrict__ Y, size_t y_zstride,
                     int Ndim, int Kdim) {
    __shared__ float Xs[2][64 * 40];   // 32 data + 8 pad per row (16|4|16|4)
    __shared__ float Ws[2][32 * 64];
    int z = blockIdx.z;
    const float* Xb = X + (size_t)(z % x_mod) * x_zstride;
    const float* Wb = W + (size_t)z * w_zstride;
    const float* bb = bias + (size_t)z * b_zstride;
    float* Yb = Y + (size_t)z * y_zstride;
    int m0 = blockIdx.y * 64, n0 = blockIdx.x * 64;
    int tid = threadIdx.x;
    int wv = tid >> 5, lane = tid & 31;
    int wm = wv >> 1, wn = wv & 1;
    int lx = lane & 15, half = lane >> 4, k0 = half * 2;

    unsigned xs_lds[2] = { (unsigned)(uintptr_t)&Xs[0][0], (unsigned)(uintptr_t)&Xs[1][0] };
    unsigned ws_lds[2] = { (unsigned)(uintptr_t)&Ws[0][0], (unsigned)(uintptr_t)&Ws[1][0] };

    int nk = Kdim / 32;
    if (tid < 32) {
        tdm_load_2d(xs_lds[0], Xb + (size_t)m0 * Kdim, 32, 64, (unsigned)Kdim,
                    TDM_F32_PAD16_4);
    } else if (tid < 64) {
        tdm_load_2d(ws_lds[0], Wb + n0, 64, 32, (unsigned)Ndim, TDM_F32);
    }

    v8f c0 = {}; v8f c1 = {};
    for (int i = 0; i < nk; ++i) {
        int cur = i & 1;
        if (i + 1 < nk) {   // prefetch next stage into the other buffer
            int kb = (i + 1) * 32;
            if (tid < 32) {
                tdm_load_2d(xs_lds[cur ^ 1], Xb + (size_t)m0 * Kdim + kb, 32, 64,
                            (unsigned)Kdim, TDM_F32_PAD16_4);
            } else if (tid < 64) {
                tdm_load_2d(ws_lds[cur ^ 1], Wb + (size_t)kb * Ndim + n0, 64, 32,
                            (unsigned)Ndim, TDM_F32);
            }
            __builtin_amdgcn_s_wait_tensorcnt((short)1);  // stage i complete
        } else {
            __builtin_amdgcn_s_wait_tensorcnt((short)0);
        }
        __syncthreads();    // publish staged tiles to all waves
        const float* Xc = Xs[cur];
        const float* Wc = Ws[cur];
        int Ma = (wm * 16 + lx) * 40;
#pragma unroll
        for (int kk = 0; kk < 32; kk += 4) {
            int col = kk + k0 + ((kk >> 4) << 2);   // skip LDS pad per 16-block
            v2f a; a.x = Xc[Ma + col]; a.y = Xc[Ma + col + 1];
            int r0 = (kk + k0) * 64 + wn * 32 + lx;
            v2f b0; b0.x = Wc[r0];      b0.y = Wc[r0 + 64];
            v2f b1; b1.x = Wc[r0 + 16]; b1.y = Wc[r0 + 80];
            c0 = wmma_f32(a, b0, c0);
            c1 = wmma_f32(a, b1, c1);
        }
        __syncthreads();    // all reads done before buffer reuse at stage i+2
    }
    int Nc = n0 + wn * 32 + lx;
    float bias0 = bb[Nc];
    float bias1 = bb[Nc + 16];
#pragma unroll
    for (int v = 0; v < 8; ++v) {
        int row = m0 + wm * 16 + v + half * 8;
        Yb[(size_t)row * Ndim + Nc]      = c0[v] + bias0;
        Yb[(size_t)row * Ndim + Nc + 16] = c1[v] + bias1;
    }
}

// ---------------------------------------------------------------------------
// Kernel 4: flash attention, fp32 WMMA. Block = 4 waves, 64 Q rows per block.
// TDM staging: wave0 DMAs Q (once) + K tiles, wave1 DMAs V tiles, K/V double
// buffered. TENSORcnt completes in-order per wave, so wait<=1 always means
// the current buffer (and Q on the first pass) is resident.
// ---------------------------------------------------------------------------
__global__ __launch_bounds__(128)
void flash_attn_kernel(const float* __restrict__ qkv, float* __restrict__ out) {
    __shared__ float Qs[64 * 68];       // stride 68: conflict-free column reads
    __shared__ float Ks[2][16 * 68];
    __shared__ float Vs[2][16 * 68];
    __shared__ float Ps[4][16 * 20];    // per-wave P staging (D-layout -> A-layout)
    int bh = blockIdx.y;
    int b = bh / N_HEADS, h = bh % N_HEADS;
    int qm0 = blockIdx.x * 64;
    const size_t plane = (size_t)T_ * D_MODEL;
    const float* Q = qkv + (size_t)(0 * B_ + b) * plane;
    const float* K = qkv + (size_t)(1 * B_ + b) * plane;
    const float* V = qkv + (size_t)(2 * B_ + b) * plane;
    int tid = threadIdx.x, wv = tid >> 5, lane = tid & 31;
    int lx = lane & 15, half = lane >> 4, k0 = half * 2;
    int hoff = h * HD;

    unsigned qs_lds    = (unsigned)(uintptr_t)&Qs[0];
    unsigned ks_lds[2] = { (unsigned)(uintptr_t)&Ks[0][0], (unsigned)(uintptr_t)&Ks[1][0] };
    unsigned vs_lds[2] = { (unsigned)(uintptr_t)&Vs[0][0], (unsigned)(uintptr_t)&Vs[1][0] };

    if (tid < 32) {
        tdm_load_2d(qs_lds, Q + (size_t)qm0 * D_MODEL + hoff, 64, 64,
                    (unsigned)D_MODEL, TDM_F32_PAD64_4);
        tdm_load_2d(ks_lds[0], K + hoff, 64, 16, (unsigned)D_MODEL, TDM_F32_PAD64_4);
    } else if (tid < 64) {
        tdm_load_2d(vs_lds[0], V + hoff, 64, 16, (unsigned)D_MODEL, TDM_F32_PAD64_4);
    }

    v8f o0 = {}, o1 = {}, o2 = {}, o3 = {};
    float mrun[8], lrun[8];
#pragma unroll
    for (int v = 0; v < 8; ++v) { mrun[v] = -INFINITY; lrun[v] = 0.f; }
    float* Pw = Ps[wv];
    int Ma = (wv * 16 + lx) * 68;
    const int nt = T_ / 16;

    for (int j = 0; j < nt; ++j) {
        int cur = j & 1;
        if (j + 1 < nt) {   // prefetch next K/V tile into the other buffer
            const size_t roff = (size_t)(j + 1) * 16 * D_MODEL + hoff;
            if (tid < 32) {
                tdm_load_2d(ks_lds[cur ^ 1], K + roff, 64, 16, (unsigned)D_MODEL,
                            TDM_F32_PAD64_4);
            } else if (tid < 64) {
                tdm_load_2d(vs_lds[cur ^ 1], V + roff, 64, 16, (unsigned)D_MODEL,
                            TDM_F32_PAD64_4);
            }
            __builtin_amdgcn_s_wait_tensorcnt((short)1);
        } else {
            __builtin_amdgcn_s_wait_tensorcnt((short)0);
        }
        __syncthreads();    // publish K/V(cur) (and Q on first pass)
        const float* Kc = Ks[cur];
        const float* Vc = Vs[cur];
        // S(16x16) = Q_tile @ K_tile^T   (16 micro-K WMMAs over head dim 64)
        v8f s = {};
#pragma unroll
        for (int kk = 0; kk < HD; kk += 4) {
            v2f a;  a.x  = Qs[Ma + kk + k0];       a.y  = Qs[Ma + kk + k0 + 1];
            v2f bf; bf.x = Kc[lx * 68 + kk + k0];  bf.y = Kc[lx * 68 + kk + k0 + 1];
            s = wmma_f32(a, bf, s);
        }
        // online softmax: C-fragment VGPR v <-> row (v + half*8); row stats are
        // 16-lane-half reductions (xor 1,2,4,8 stays within the half in wave32)
#pragma unroll
        for (int v = 0; v < 8; ++v) {
            float x = s[v] * 0.125f;   // 1/sqrt(64)
            float mx = x;
            mx = fmaxf(mx, __shfl_xor(mx, 1));
            mx = fmaxf(mx, __shfl_xor(mx, 2));
            mx = fmaxf(mx, __shfl_xor(mx, 4));
            mx = fmaxf(mx, __shfl_xor(mx, 8));
            float mnew = fmaxf(mrun[v], mx);
            float p = __expf(x - mnew);
            float rs = p;
            rs += __shfl_xor(rs, 1);
            rs += __shfl_xor(rs, 2);
            rs += __shfl_xor(rs, 4);
            rs += __shfl_xor(rs, 8);
            float alpha = __expf(mrun[v] - mnew);
            lrun[v] = lrun[v] * alpha + rs;
            mrun[v] = mnew;
            o0[v] *= alpha; o1[v] *= alpha; o2[v] *= alpha; o3[v] *= alpha;
            Pw[(v + half * 8) * 20 + lx] = p;   // D-layout -> row-major staging
        }
        __syncthreads();
        // O(16x64) += P(16x16) @ V_tile(16x64)
#pragma unroll
        for (int kk = 0; kk < 16; kk += 4) {
            v2f a; a.x = Pw[lx * 20 + kk + k0]; a.y = Pw[lx * 20 + kk + k0 + 1];
            int r0 = (kk + k0) * 68 + lx, r1 = (kk + k0 + 1) * 68 + lx;
            v2f bf;
            bf.x = Vc[r0];      bf.y = Vc[r1];      o0 = wmma_f32(a, bf, o0);
            bf.x = Vc[r0 + 16]; bf.y = Vc[r1 + 16]; o1 = wmma_f32(a, bf, o1);
            bf.x = Vc[r0 + 32]; bf.y = Vc[r1 + 32]; o2 = wmma_f32(a, bf, o2);
            bf.x = Vc[r0 + 48]; bf.y = Vc[r1 + 48]; o3 = wmma_f32(a, bf, o3);
        }
        __syncthreads();    // K/V(cur) reads done before stage j+2 overwrites
    }
#pragma unroll
    for (int v = 0; v < 8; ++v) {
        float inv = 1.f / lrun[v];
        int row = qm0 + wv * 16 + v + half * 8;
        float* dst = out + (size_t)b * plane + (size_t)row * D_MODEL + hoff + lx;
        dst[0]  = o0[v] * inv;
        dst[16] = o1[v] * inv;
        dst[32] = o2[v] * inv;
        dst[48] = o3[v] * inv;
    }
}

// ---------------------------------------------------------------------------
extern "C" void kernel_launch(void* const* d_in, const int* in_sizes, int n_in,
                              void* d_out, int out_size, void* d_ws, size_t ws_size,
                              hipStream_t stream) {
    (void)in_sizes; (void)n_in; (void)out_size; (void)ws_size;
    const float* hs  = (const float*)d_in[0];
    const float* Wq  = (const float*)d_in[1];
    const float* bq  = (const float*)d_in[2];
    const float* Wk  = (const float*)d_in[3];
    const float* bk  = (const float*)d_in[4];
    const float* Wv  = (const float*)d_in[5];
    const float* bv  = (const float*)d_in[6];
    const float* Wg1 = (const float*)d_in[7];
    const float* bg1 = (const float*)d_in[8];
    const float* Wg2 = (const float*)d_in[9];
    const float* bg2 = (const float*)d_in[10];
    const float* Wo  = (const float*)d_in[11];
    const float* bo  = (const float*)d_in[12];
    float* out = (float*)d_out;

    const size_t CC = (size_t)D_MODEL * D_MODEL;
    const size_t TC = (size_t)T_ * D_MODEL;
    float* wsf   = (float*)d_ws;
    float* wgate = wsf;                         // B*E (padded to 64)
    float* Wc    = wsf + 64;                    // 3*B*C*C
    float* bc    = Wc + 3 * B_ * CC;            // 3*B*C
    float* qkvb  = bc + 3 * B_ * D_MODEL;       // 3*B*T*C  ([p][b][T][C])
    float* att   = qkvb + 3 * B_ * TC;          // B*T*C

    gate_kernel<<<1, 256, 0, stream>>>(hs, Wg1, bg1, Wg2, bg2, wgate);
    combine_w_kernel<<<1024, 256, 0, stream>>>(Wq, Wk, Wv, wgate, Wc);
    combine_b_kernel<<<8, 256, 0, stream>>>(bq, bk, bv, wgate, bc);

    // QKV: z = p*B + b ; X selected by b = z % B
    dim3 g1(D_MODEL / 64, T_ / 64, 3 * B_);
    gemm_moe_kernel<<<g1, 256, 0, stream>>>(hs, TC, B_, Wc, CC, bc, D_MODEL,
                                            qkvb, TC, D_MODEL, D_MODEL);

    dim3 g2(T_ / 64, B_ * N_HEADS);
    flash_attn_kernel<<<g2, 128, 0, stream>>>(qkvb, att);

    // output projection: shared Wo/bo (z strides 0)
    dim3 g3(D_MODEL / 64, T_ / 64, B_);
    gemm_moe_kernel<<<g3, 256, 0, stream>>>(att, TC, 1 << 30, Wo, 0, bo, 0,
                                            out, TC, D_MODEL, D_MODEL);
}